// get_candidate_layer_52132313038912
// MI455X (gfx1250) — compile-verified
//
#include <hip/hip_runtime.h>
#include <hip/hip_bf16.h>
#include <stdint.h>

// ---------------- problem constants (from reference setup) ----------------
#define IMG_W   768.0f
#define IMG_H   432.0f
#define BATCH   8
#define NANCH   1000000
#define NB      4096        // histogram buckets for radix-select
#define CAP     16384       // max candidates kept per batch (>= K + bucket width)
#define SORT_N  16384       // bitonic sort size (power of two, 128KB LDS of u64)

typedef float    v4f __attribute__((ext_vector_type(4)));
typedef uint32_t u32;
typedef uint64_t u64;
typedef u32 v4u __attribute__((ext_vector_type(4)));
typedef int  v8i __attribute__((ext_vector_type(8)));
typedef int  v4i __attribute__((ext_vector_type(4)));

// ---- CDNA5 data-mover availability probes (warnings surface in stderr) ----
#if defined(__HIP_DEVICE_COMPILE__) && defined(__gfx1250__)
  #if __has_builtin(__builtin_amdgcn_tensor_load_to_lds) && \
      __has_builtin(__builtin_amdgcn_s_wait_tensorcnt)
    #define USE_TDM 1
  #else
    #define USE_TDM 0
    #warning "CDNA5 probe: __builtin_amdgcn_tensor_load_to_lds NOT available on this toolchain"
  #endif
  #if __has_builtin(__builtin_amdgcn_global_load_async_to_lds_b64) && \
      __has_builtin(__builtin_amdgcn_s_wait_asynccnt)
    #define USE_ASYNC_LDS 1
  #else
    #define USE_ASYNC_LDS 0
    #warning "CDNA5 probe: __builtin_amdgcn_global_load_async_to_lds_b64 NOT available on this toolchain"
  #endif
#else
  #define USE_TDM 0
  #define USE_ASYNC_LDS 0
#endif

__device__ __forceinline__ v4f clip_box(v4f b) {
    float cx = b.x, cy = b.y, w = b.z, h = b.w;
    float x1 = fminf(fmaxf(cx - w * 0.5f, 0.0f), IMG_W);
    float x2 = fminf(fmaxf(cx + w * 0.5f, 0.0f), IMG_W);
    float y1 = fminf(fmaxf(cy - h * 0.5f, 0.0f), IMG_H);
    float y2 = fminf(fmaxf(cy + h * 0.5f, 0.0f), IMG_H);
    v4f r;
    r.z = x2 - x1;            // nw
    r.w = y2 - y1;            // nh
    r.x = x1 + r.z * 0.5f;    // nx
    r.y = y1 + r.w * 0.5f;    // ny
    return r;
}

// ---------------- kernel 0: zero the scratch counters ----------------
__global__ void k_init(u32* __restrict__ hist, u32* __restrict__ thresh,
                       u32* __restrict__ ccount) {
    int i = blockIdx.x * blockDim.x + threadIdx.x;
    if (i < BATCH * NB) hist[i] = 0u;
    if (i < BATCH * 2)  thresh[i] = 0u;
    if (i < BATCH)      ccount[i] = 0u;
}

// ---------------- kernel 1: clip + mask + score histogram (LDS-staged) ----------------
__global__ __launch_bounds__(256) void k_hist(const float* __restrict__ scores,
                                              const float* __restrict__ rps,
                                              float* __restrict__ S,
                                              u32* __restrict__ hist) {
    __shared__ u32 lh[NB];
    const int b = blockIdx.y;
    for (int i = threadIdx.x; i < NB; i += 256) lh[i] = 0u;
    __syncthreads();

    const int base = blockIdx.x * 4096;
    const int end  = (base + 4096 < NANCH) ? base + 4096 : NANCH;
    const v4f*   rp = (const v4f*)rps + (size_t)b * NANCH;
    const float* sp = scores + (size_t)b * NANCH;
    float*       Sp = S + (size_t)b * NANCH;

    for (int i = base + threadIdx.x; i < end; i += 256) {
        __builtin_prefetch(rp + i + 2048, 0, 0);        // global_prefetch_b8, speculative
        v4f box = __builtin_nontemporal_load(rp + i);   // 160MB stream: NT hint
        v4f r   = clip_box(box);
        float sc = __builtin_nontemporal_load(sp + i);
        float s  = (r.z > 16.0f && r.w > 16.0f) ? sc : 0.0f;
        Sp[i] = s;                                  // RT store -> stays in 192MB L2
        int bk = (int)(s * (float)NB);
        bk = bk > NB - 1 ? NB - 1 : bk;             // s in [0,1): bk in [0,4095]
        atomicAdd(&lh[bk], 1u);
    }
    __syncthreads();
    for (int i = threadIdx.x; i < NB; i += 256) {
        u32 c = lh[i];
        if (c) atomicAdd(&hist[b * NB + i], c);
    }
}

// ---------------- kernel 2: find K-th-largest threshold bucket ----------------
__global__ __launch_bounds__(256) void k_thresh(const u32* __restrict__ hist,
                                                const int* __restrict__ pK,
                                                u32* __restrict__ thresh) {
    __shared__ u32 psum[256];
    const int b = blockIdx.x;
    const int t = threadIdx.x;
    const u32 Ksel = (u32)pK[0];
    const int hi = NB - 1 - (t << 4);      // chunk t covers buckets [hi-15, hi], high->low
    u32 sum = 0;
    for (int u = 0; u < 16; ++u) sum += hist[b * NB + hi - u];
    psum[t] = sum;
    __syncthreads();
    // Hillis-Steele inclusive scan over chunks (high buckets first)
    for (int off = 1; off < 256; off <<= 1) {
        u32 v = (t >= off) ? psum[t - off] : 0u;
        __syncthreads();
        psum[t] += v;
        __syncthreads();
    }
    u32 incl = psum[t];
    u32 excl = incl - sum;
    if (excl < Ksel && incl >= Ksel) {     // unique crossing chunk
        u32 acc = excl;
        for (int u = 0; u < 16; ++u) {
            u32 c = hist[b * NB + hi - u];
            if (acc + c >= Ksel) {
                thresh[b * 2 + 0] = (u32)(hi - u);  // threshold bucket T
                thresh[b * 2 + 1] = acc;            // count strictly above T
                break;
            }
            acc += c;
        }
    }
}

// ---------------- kernel 3: compact candidates (score bucket >= T) ----------------
__global__ __launch_bounds__(256) void k_compact(const float* __restrict__ S,
                                                 const u32* __restrict__ thresh,
                                                 u64* __restrict__ cand,
                                                 u32* __restrict__ ccount) {
    const int b = blockIdx.y;
    const u32 T = thresh[b * 2];
    const int base = blockIdx.x * 4096;
    const int end  = (base + 4096 < NANCH) ? base + 4096 : NANCH;
    const float* Sp = S + (size_t)b * NANCH;
    for (int i = base + threadIdx.x; i < end; i += 256) {
        float s = Sp[i];
        int bk = (int)(s * (float)NB);
        bk = bk > NB - 1 ? NB - 1 : bk;
        if ((u32)bk >= T) {
            u32 pos = atomicAdd(&ccount[b], 1u);
            if (pos < CAP) {
                // key: score bits (positive float -> order-preserving) | inverted index
                cand[(size_t)b * CAP + pos] =
                    ((u64)__float_as_uint(s) << 32) | (u64)(0xFFFFFFFFu - (u32)i);
            }
        }
    }
}

// ---------------- kernel 4: TDM-load candidates to LDS, bitonic sort, emit top-K ----------------
__global__ __launch_bounds__(1024) void k_sort_out(const float* __restrict__ rps,
                                                   const u64* __restrict__ cand,
                                                   const u32* __restrict__ ccount,
                                                   const int* __restrict__ pK,
                                                   float* __restrict__ out_rois,
                                                   float* __restrict__ out_s) {
    extern __shared__ u64 keys[];          // SORT_N entries (128KB of 320KB WGP LDS)
    const int b = blockIdx.x;
    const int tid = threadIdx.x;
    const int Ksel = pK[0];
    u32 M = ccount[b];
    if (M > CAP) M = CAP;
#if defined(__HIP_DEVICE_COMPILE__)
    M = (u32)__builtin_amdgcn_readfirstlane((int)M);   // make wave-uniform (SGPR)
#endif
    // pad tail with 0 (real keys always have nonzero low word -> 0 is the unique sentinel)
    for (int i = tid; i < SORT_N; i += 1024)
        if ((u32)i >= M) keys[i] = 0ull;

#if USE_TDM
    // One wave issues the Tensor Data Mover DMA: global cand[b][0..M) -> LDS.
    if (tid < 32) {
        unsigned long long ga = (unsigned long long)(uintptr_t)(cand + (size_t)b * CAP);
        v4u g0;
        g0.x = 1u;                                         // count=1, user descriptor
        g0.y = (u32)__builtin_amdgcn_groupstaticsize();    // dynamic-LDS base offset
        g0.z = (u32)(ga & 0xFFFFFFFFull);                  // global_addr[31:0]
        g0.w = (u32)((ga >> 32) & 0x01FFFFFFull) | (2u << 30);  // addr[56:32] | type=2
        v8i g1;
        g1[0] = (int)(3u << 16);                           // data_size = 8 bytes
        g1[1] = (int)((M & 0xFFFFu) << 16);                // tensor_dim0[15:0]
        g1[2] = (int)(((M >> 16) & 0xFFFFu) | (1u << 16)); // tensor_dim0[31:16], tensor_dim1=1
        g1[3] = (int)((M & 0xFFFFu) << 16);                // tile_dim0 = M (1D tile)
        g1[4] = 0;                                         // tile_dim1/2 unused
        g1[5] = (int)M;                                    // tensor_dim0_stride
        g1[6] = 0;
        g1[7] = 0;
        v4i z4 = {0, 0, 0, 0};
#if __clang_major__ >= 23
        v8i z8 = {0, 0, 0, 0, 0, 0, 0, 0};
        __builtin_amdgcn_tensor_load_to_lds(g0, g1, z4, z4, z8, 0);
#else
        __builtin_amdgcn_tensor_load_to_lds(g0, g1, z4, z4, 0);
#endif
        __builtin_amdgcn_s_wait_tensorcnt(0);
    }
#elif USE_ASYNC_LDS
    // Async LDS copy path (ASYNCcnt-tracked), 8B per lane per iteration.
    {
        __attribute__((address_space(3))) u64* lk =
            (__attribute__((address_space(3))) u64*)keys;
        for (u32 i = (u32)tid; i < M; i += 1024)
            __builtin_amdgcn_global_load_async_to_lds_b64(
                (const __attribute__((address_space(1))) void*)(cand + (size_t)b * CAP + i),
                (__attribute__((address_space(3))) void*)(lk + i), 0, 0);
        __builtin_amdgcn_s_wait_asynccnt(0);
    }
#else
    for (u32 i = (u32)tid; i < M; i += 1024)
        keys[i] = cand[(size_t)b * CAP + i];
#endif
    __syncthreads();

    // bitonic sort, descending (ties: larger ~idx first => ascending index, stable-argsort match)
    for (u32 kk = 2; kk <= SORT_N; kk <<= 1) {
        for (u32 j = kk >> 1; j > 0; j >>= 1) {
            for (u32 i = (u32)tid; i < SORT_N; i += 1024) {
                u32 ixj = i ^ j;
                if (ixj > i) {
                    u64 a = keys[i], c = keys[ixj];
                    bool up = ((i & kk) == 0);
                    if ((a < c) == up) { keys[i] = c; keys[ixj] = a; }
                }
            }
            __syncthreads();
        }
    }

    // emit top-K: recompute clipped roi from rps at selected indices
    const v4f* rp = (const v4f*)rps + (size_t)b * NANCH;
    v4f* orow = (v4f*)out_rois + (size_t)b * Ksel;
    for (int q = tid; q < Ksel; q += 1024) {
        u64 key = keys[q];
        float s = 0.0f;
        v4f r = {0.0f, 0.0f, 0.0f, 0.0f};
        if (key != 0ull) {
            u32 idx = 0xFFFFFFFFu - (u32)(key & 0xFFFFFFFFull);
            s = __uint_as_float((u32)(key >> 32));
            r = clip_box(rp[idx]);
        }
        orow[q] = r;                                 // single b128 store
        out_s[(size_t)b * Ksel + q] = s;
    }
}

// ---------------- launch ----------------
extern "C" void kernel_launch(void* const* d_in, const int* in_sizes, int n_in,
                              void* d_out, int out_size, void* d_ws, size_t ws_size,
                              hipStream_t stream) {
    const float* scores = (const float*)d_in[0];   // [B, N, 1]
    const float* rps    = (const float*)d_in[1];   // [B, N, 4]
    const int*   pK     = (const int*)d_in[2];     // scalar K (=12000)

    // workspace layout (~33.2 MB): S | hist | thresh | ccount | cand
    char* ws = (char*)d_ws;
    float* S = (float*)ws;
    size_t off = (size_t)BATCH * NANCH * sizeof(float);          // 32 MB
    u32* hist   = (u32*)(ws + off);  off += (size_t)BATCH * NB * sizeof(u32);
    u32* thresh = (u32*)(ws + off);  off += (size_t)BATCH * 2 * sizeof(u32);
    u32* ccount = (u32*)(ws + off);  off += (size_t)BATCH * sizeof(u32);
    off = (off + 7) & ~(size_t)7;
    u64* cand   = (u64*)(ws + off);                               // 1 MB

    // output split: rois [B,K,4] then scores [B,K,1]
    int Kout = out_size / (BATCH * 5);
    float* out_rois = (float*)d_out;
    float* out_s    = out_rois + (size_t)BATCH * Kout * 4;

    k_init<<<(BATCH * NB + 255) / 256, 256, 0, stream>>>(hist, thresh, ccount);
    dim3 grid((NANCH + 4095) / 4096, BATCH);
    k_hist   <<<grid, 256, 0, stream>>>(scores, rps, S, hist);
    k_thresh <<<BATCH, 256, 0, stream>>>(hist, pK, thresh);
    k_compact<<<grid, 256, 0, stream>>>(S, thresh, cand, ccount);
    k_sort_out<<<BATCH, 1024, SORT_N * sizeof(u64), stream>>>(rps, cand, ccount, pK,
                                                              out_rois, out_s);
}